// Graph_Sage_GNN_609885356389
// MI455X (gfx1250) — compile-verified
//
#include <hip/hip_runtime.h>

// ---------------------------------------------------------------------------
// GraphSAGE 2-layer GNN for MI455X (gfx1250).
//   per layer: h = relu(x @ Wp.T + bp)
//              agg = segment_mean(h[src] -> dst)
//              out = agg @ Wl.T + bl + x @ Wr.T
// Dense GEMMs use V_WMMA_F32_16X16X4_F32 (full f32 precision, ~10 GF total so
// matrix-core f32 rate is ample). Edge aggregation = wave-per-edge float4
// gather + global_atomic_add_f32 scatter (~1.6 GB total, HBM/atomic bound).
// Mean division is folded into the A-operand load of the following GEMM.
// ---------------------------------------------------------------------------

typedef __attribute__((ext_vector_type(2))) float v2f;
typedef __attribute__((ext_vector_type(8))) float v8f;

#define D_FEAT 128
#define LDS_STRIDE 132   // padded row stride (floats): 16B-aligned rows, banks 4n -> conflict-free

// ---------------------------------------------------------------------------
// Fused GEMM:  Out = act( A1*rowscale @ W1.T  [+ A2 @ W2.T]  + bias )
//   A1,A2 : [M,128] row-major       W1,W2 : [128,128] row-major (W.T applied)
//   rowcnt: optional per-row divisor (mean aggregation), scale = 1/max(cnt,1)
// Block = 256 threads = 8 waves; each wave computes a 16x128 output tile with
// 8 v8f accumulators; weights staged into LDS once per pass.
// Requires M % 16 == 0 (true: 50000 = 16*3125).
// ---------------------------------------------------------------------------
__global__ __launch_bounds__(256) void sage_gemm_kernel(
    const float* __restrict__ A1, const float* __restrict__ W1,
    const float* __restrict__ bias,
    const float* __restrict__ A2, const float* __restrict__ W2,
    const float* __restrict__ rowcnt,
    float* __restrict__ Out, int M, int do_relu)
{
    __shared__ float Wlds[D_FEAT * LDS_STRIDE];

    const int tid   = threadIdx.x;
    const int wid   = tid >> 5;
    const int lane  = tid & 31;
    const int l16   = lane & 15;
    const int khalf = (lane >> 4) << 1;               // 0 for lanes 0-15, 2 for 16-31
    const long m0   = ((long)blockIdx.x * 8 + wid) * 16;
    const bool active = (m0 < (long)M);               // wave-uniform predicate
    const long row  = m0 + l16;                       // A-operand row for this lane

    v8f acc[8];
#pragma unroll
    for (int i = 0; i < 8; ++i)
#pragma unroll
        for (int c = 0; c < 8; ++c) acc[i][c] = 0.0f;

    float rs = 1.0f;
    if (active && rowcnt) rs = 1.0f / fmaxf(rowcnt[row], 1.0f);

    // stage one 128x128 f32 weight matrix into padded LDS
    auto stage = [&](const float* __restrict__ W) {
        const int r  = tid >> 1;
        const int hf = (tid & 1) << 6;                // 0 or 64 floats
        const float* src = W + r * D_FEAT + hf;
        float* dst = &Wlds[r * LDS_STRIDE + hf];
#pragma unroll
        for (int i = 0; i < 16; ++i)
            *(float4*)(dst + 4 * i) = *(const float4*)(src + 4 * i);
    };

    // one accumulation pass: acc += (A * scale) @ Wlds.T
    auto pass = [&](const float* __restrict__ A, float scale) {
        for (int k0 = 0; k0 < D_FEAT; k0 += 4) {
            v2f a = *(const v2f*)(A + row * D_FEAT + k0 + khalf);  // A[m][k..k+1]
            a = a * scale;
#pragma unroll
            for (int nt = 0; nt < 8; ++nt) {
                const int n = nt * 16 + l16;
                // B[k][n] = W[n][k]: contiguous v2f from LDS row n
                v2f b = *(const v2f*)(&Wlds[n * LDS_STRIDE + k0 + khalf]);
                acc[nt] = __builtin_amdgcn_wmma_f32_16x16x4_f32(
                    false, a, false, b, (short)0, acc[nt], false, false);
            }
        }
    };

    stage(W1);
    __syncthreads();
    if (active) pass(A1, rs);

    if (A2) {
        __syncthreads();
        stage(W2);
        __syncthreads();
        if (active) pass(A2, 1.0f);
    }

    if (active) {
        // C/D layout: VGPR r -> row m0+r (lanes 0-15) or m0+8+r (lanes 16-31), col = l16
        const long rbase = m0 + ((lane >> 4) << 3);
#pragma unroll
        for (int nt = 0; nt < 8; ++nt) {
            const float bval = bias[nt * 16 + l16];
#pragma unroll
            for (int r = 0; r < 8; ++r) {
                float v = acc[nt][r] + bval;
                if (do_relu) v = fmaxf(v, 0.0f);
                Out[(rbase + r) * D_FEAT + nt * 16 + l16] = v;
            }
        }
    }
}

// ---------------------------------------------------------------------------
// Edge aggregation: one wave per edge. lane l moves features [4l, 4l+4):
//   agg[dst] += h[src]   (global_atomic_add_f32)
// lane 0 optionally accumulates the in-degree count (float, exact to 2^24).
// ---------------------------------------------------------------------------
__global__ __launch_bounds__(256) void sage_edge_kernel(
    const float* __restrict__ h, const int* __restrict__ eidx,
    float* __restrict__ agg, float* __restrict__ cnt, int E)
{
    const int wave = (int)((blockIdx.x * (unsigned)blockDim.x + threadIdx.x) >> 5);
    const int lane = threadIdx.x & 31;
    if (wave >= E) return;

    const long src = (long)eidx[wave];        // row 0 of edge_index
    const long dst = (long)eidx[E + wave];    // row 1 of edge_index

    const float4 v = *(const float4*)(h + src * D_FEAT + lane * 4);
    float* p = agg + dst * D_FEAT + lane * 4;
    unsafeAtomicAdd(p + 0, v.x);
    unsafeAtomicAdd(p + 1, v.y);
    unsafeAtomicAdd(p + 2, v.z);
    unsafeAtomicAdd(p + 3, v.w);

    if (cnt && lane == 0) unsafeAtomicAdd(cnt + dst, 1.0f);
}

// ---------------------------------------------------------------------------
extern "C" void kernel_launch(void* const* d_in, const int* in_sizes, int n_in,
                              void* d_out, int out_size, void* d_ws, size_t ws_size,
                              hipStream_t stream)
{
    (void)n_in; (void)out_size; (void)ws_size;

    const float* x   = (const float*)d_in[0];
    const int*   eix = (const int*)d_in[1];
    const float* Wp1 = (const float*)d_in[2];
    const float* bp1 = (const float*)d_in[3];
    const float* Wl1 = (const float*)d_in[4];
    const float* bl1 = (const float*)d_in[5];
    const float* Wr1 = (const float*)d_in[6];
    const float* Wp2 = (const float*)d_in[7];
    const float* bp2 = (const float*)d_in[8];
    const float* Wl2 = (const float*)d_in[9];
    const float* bl2 = (const float*)d_in[10];
    const float* Wr2 = (const float*)d_in[11];
    float* out = (float*)d_out;

    const int M = in_sizes[0] / D_FEAT;   // 50000 nodes
    const int E = in_sizes[1] / 2;        // 800000 edges

    // workspace layout: h | agg | l1 | cnt  (~77 MB)
    char* ws = (char*)d_ws;
    const size_t featBytes = (size_t)M * D_FEAT * sizeof(float);
    float* bufH   = (float*)(ws);
    float* bufAgg = (float*)(ws + featBytes);
    float* bufL1  = (float*)(ws + 2 * featBytes);
    float* cnt    = (float*)(ws + 3 * featBytes);

    const int waveGroups = (M + 15) / 16;            // 16-row tiles
    const int gemmBlocks = (waveGroups + 7) / 8;     // 8 waves per block
    const int edgeBlocks = (E + 7) / 8;              // 8 edges per 256-thread block

    // ---- layer 1 ----
    hipMemsetAsync(bufAgg, 0, featBytes, stream);
    hipMemsetAsync(cnt, 0, (size_t)M * sizeof(float), stream);
    sage_gemm_kernel<<<gemmBlocks, 256, 0, stream>>>(
        x, Wp1, bp1, nullptr, nullptr, nullptr, bufH, M, 1);
    sage_edge_kernel<<<edgeBlocks, 256, 0, stream>>>(bufH, eix, bufAgg, cnt, E);
    sage_gemm_kernel<<<gemmBlocks, 256, 0, stream>>>(
        bufAgg, Wl1, bl1, x, Wr1, cnt, bufL1, M, 1);

    // ---- layer 2 (degree counts reused) ----
    hipMemsetAsync(bufAgg, 0, featBytes, stream);
    sage_gemm_kernel<<<gemmBlocks, 256, 0, stream>>>(
        bufL1, Wp2, bp2, nullptr, nullptr, nullptr, bufH, M, 1);
    sage_edge_kernel<<<edgeBlocks, 256, 0, stream>>>(bufH, eix, bufAgg, nullptr, E);
    sage_gemm_kernel<<<gemmBlocks, 256, 0, stream>>>(
        bufAgg, Wl2, bl2, bufL1, Wr2, cnt, out, M, 0);
}